// NonLocalBlock_67422396613171
// MI455X (gfx1250) — compile-verified
//
#include <hip/hip_runtime.h>
#include <math.h>

// ---------------------------------------------------------------------------
// NonLocalBlock for MI455X (gfx1250, wave32).
// Attention GEMMs run on the fp32 matrix pipe: V_WMMA_F32_16X16X4_F32.
// Flash-style chunked softmax in LDS avoids materializing the 340MB score
// tensor (HBM-roofline dominated otherwise).
// ---------------------------------------------------------------------------

typedef float v2f __attribute__((ext_vector_type(2)));
typedef float v8f __attribute__((ext_vector_type(8)));

#define Bn   4
#define Cc   64
#define ICc  32
#define Hh   96
#define Ww   96
#define Np   (Hh * Ww)   // 9216 queries
#define HP   48
#define Mp   (HP * HP)   // 2304 keys (after 2x2 maxpool)
#define CHUNK 576        // keys per LDS softmax chunk (16 x 576 fp32 = 36 KB)
#define BN_EPSf 1e-5f
#define NEG_BIG (-3.4e38f)

__device__ __forceinline__ v8f wmma_f32(v2f a, v2f b, v8f c) {
  // D = A(16x4) * B(4x16) + C(16x16), fp32 throughout.
  return __builtin_amdgcn_wmma_f32_16x16x4_f32(
      /*neg_a=*/false, a, /*neg_b=*/false, b,
      /*c_mod=*/(short)0, c, /*reuse_a=*/false, /*reuse_b=*/false);
}

// --------------------------- kernel 0: zero BN stats ------------------------
__global__ void k_zero_sums(float* __restrict__ sums) {
  if (threadIdx.x < 128) sums[threadIdx.x] = 0.0f;
}

// --------------------------- kernel 1: theta projection ---------------------
// theta[b][o][n] = sum_c w_theta[o][c] * x[b][c][n]
__global__ __launch_bounds__(256)
void k_proj_theta(const float* __restrict__ x, const float* __restrict__ wt,
                  float* __restrict__ theta) {
  int idx = blockIdx.x * blockDim.x + threadIdx.x;
  if (idx >= Bn * Np) return;
  int b = idx / Np, n = idx % Np;
  const float* xp = x + (size_t)b * Cc * Np + n;
  float xr[Cc];
#pragma unroll 8
  for (int c = 0; c < Cc; ++c) xr[c] = xp[(size_t)c * Np];
  for (int o = 0; o < ICc; ++o) {
    float s = 0.0f;
#pragma unroll 8
    for (int c = 0; c < Cc; ++c) s = fmaf(wt[o * Cc + c], xr[c], s);
    theta[((size_t)b * ICc + o) * Np + n] = s;
  }
}

// ----------------- kernel 2: phi/g projection + fused 2x2 maxpool -----------
__global__ __launch_bounds__(256)
void k_proj_pool(const float* __restrict__ y, const float* __restrict__ wphi,
                 const float* __restrict__ wg, float* __restrict__ phi,
                 float* __restrict__ g) {
  int idx = blockIdx.x * blockDim.x + threadIdx.x;
  if (idx >= Bn * Mp) return;
  int b = idx / Mp, m = idx % Mp;
  int prow = m / HP, pcol = m % HP;
  float pmax[ICc], gmax[ICc];
#pragma unroll
  for (int o = 0; o < ICc; ++o) { pmax[o] = NEG_BIG; gmax[o] = NEG_BIG; }
  for (int p = 0; p < 4; ++p) {
    int r = 2 * prow + (p >> 1), cl = 2 * pcol + (p & 1);
    float yr[Cc];
#pragma unroll 8
    for (int c = 0; c < Cc; ++c)
      yr[c] = y[((size_t)(b * Cc + c) * Hh + r) * Ww + cl];
    for (int o = 0; o < ICc; ++o) {
      float sp = 0.0f, sg = 0.0f;
#pragma unroll 8
      for (int c = 0; c < Cc; ++c) {
        sp = fmaf(wphi[o * Cc + c], yr[c], sp);
        sg = fmaf(wg[o * Cc + c], yr[c], sg);
      }
      pmax[o] = fmaxf(pmax[o], sp);
      gmax[o] = fmaxf(gmax[o], sg);
    }
  }
#pragma unroll
  for (int o = 0; o < ICc; ++o) {
    phi[((size_t)b * ICc + o) * Mp + m] = pmax[o];
    g[((size_t)b * ICc + o) * Mp + m] = gmax[o];
  }
}

// ------------- kernel 3: WMMA flash attention (one wave = 16 queries) -------
// fxy[b][c'][n] = sum_m softmax_m( sum_c theta[b][c][n]*phi[b][c][m] ) * g[b][c'][m]
__global__ __launch_bounds__(32)
void k_attention(const float* __restrict__ theta, const float* __restrict__ phi,
                 const float* __restrict__ gbuf, float* __restrict__ fxy) {
  __shared__ float sS[16 * CHUNK];           // score / probability stripe
  __shared__ float sM[16], sL[16], sF[16];   // running max / denom / rescale

  const int b = blockIdx.y;
  const int n0 = blockIdx.x * 16;
  const int lane = threadIdx.x;
  const int half = lane >> 4;    // 0: lanes 0-15, 1: lanes 16-31
  const int l15 = lane & 15;

  // Preload A tiles: A[M=n_local][K=c] = theta[c][n0+n_local], K=32 in 8 chunks.
  // ISA A-layout (16x4 fp32): lane<16 -> K={0,1}; lane>=16 -> K={2,3}.
  const float* th = theta + (size_t)b * ICc * Np + n0;
  v2f a[8];
#pragma unroll
  for (int kc = 0; kc < 8; ++kc) {
    int c = kc * 4 + 2 * half;
    a[kc].x = th[(size_t)c * Np + l15];
    a[kc].y = th[(size_t)(c + 1) * Np + l15];
  }

  if (lane < 16) { sM[lane] = NEG_BIG; sL[lane] = 0.0f; }
  v8f acc0 = {};  // D[n_local][c'=0..15]
  v8f acc1 = {};  // D[n_local][c'=16..31]
  __syncthreads();

  const float* ph = phi + (size_t)b * ICc * Mp;
  const float* gg = gbuf + (size_t)b * ICc * Mp;

  for (int ch0 = 0; ch0 < Mp; ch0 += CHUNK) {
    // --- GEMM1: S(16 x CHUNK) = theta_tile^T * phi, straight into LDS ------
    for (int mt = 0; mt < CHUNK; mt += 16) {
      v8f s = {};
#pragma unroll
      for (int kc = 0; kc < 8; ++kc) {
        int c = kc * 4 + 2 * half;  // B-layout: VGPR0 rows K=0/K=2 per half-wave
        v2f bt;
        bt.x = ph[(size_t)c * Mp + ch0 + mt + l15];
        bt.y = ph[(size_t)(c + 1) * Mp + ch0 + mt + l15];
        s = wmma_f32(a[kc], bt, s);
      }
#pragma unroll
      for (int v = 0; v < 8; ++v)          // C-layout: D[v+8*half][l15]
        sS[(v + 8 * half) * CHUNK + mt + l15] = s[v];
    }
    __syncthreads();

    // --- online softmax over this chunk (2 lanes per query row) ------------
    {
      const int row = lane >> 1;
      const int hh = lane & 1;
      float* rs = &sS[row * CHUNK + hh * (CHUNK / 2)];
      float cmax = NEG_BIG;
      for (int j = 0; j < CHUNK / 2; ++j) cmax = fmaxf(cmax, rs[j]);
      cmax = fmaxf(cmax, __shfl_xor(cmax, 1, 32));
      float mprev = sM[row];
      float nm = fmaxf(mprev, cmax);
      float fac = __expf(mprev - nm);
      float psum = 0.0f;
      for (int j = 0; j < CHUNK / 2; ++j) {
        float pv = __expf(rs[j] - nm);
        rs[j] = pv;
        psum += pv;
      }
      psum += __shfl_xor(psum, 1, 32);
      if (hh == 0) { sL[row] = sL[row] * fac + psum; sM[row] = nm; sF[row] = fac; }
    }
    __syncthreads();

    // --- rescale running accumulators by exp(m_old - m_new) per row --------
#pragma unroll
    for (int v = 0; v < 8; ++v) {
      float f = sF[v + 8 * half];
      acc0[v] *= f;
      acc1[v] *= f;
    }

    // --- GEMM2: acc(16 x 32) += P(16 x CHUNK) * g^T(CHUNK x 32) ------------
    for (int mt = 0; mt < CHUNK; mt += 16) {
#pragma unroll
      for (int kk = 0; kk < 4; ++kk) {
        int mb = mt + kk * 4 + 2 * half;  // local key index incl. half-wave K-split
        v2f pa;                            // A-layout read of P from LDS
        pa.x = sS[l15 * CHUNK + mb];
        pa.y = sS[l15 * CHUNK + mb + 1];
        int gm = ch0 + mb;                 // B[K][j] = g[c0+j][gm+K]
        v2f g0, g1;
        g0.x = gg[(size_t)l15 * Mp + gm];
        g0.y = gg[(size_t)l15 * Mp + gm + 1];
        g1.x = gg[(size_t)(16 + l15) * Mp + gm];
        g1.y = gg[(size_t)(16 + l15) * Mp + gm + 1];
        acc0 = wmma_f32(pa, g0, acc0);
        acc1 = wmma_f32(pa, g1, acc1);
      }
    }
    __syncthreads();
  }

  // --- normalize by softmax denominator and write fxy[b][c'][n] ------------
#pragma unroll
  for (int v = 0; v < 8; ++v) {
    int n = n0 + v + 8 * half;
    float r = 1.0f / sL[v + 8 * half];
    fxy[((size_t)b * ICc + l15) * Np + n] = acc0[v] * r;
    fxy[((size_t)b * ICc + 16 + l15) * Np + n] = acc1[v] * r;
  }
}

// --------- kernel 4: W 1x1 conv + per-channel sum/sumsq for BatchNorm -------
__global__ __launch_bounds__(256)
void k_wxy_stats(const float* __restrict__ fxy, const float* __restrict__ ww,
                 float* __restrict__ wxy, float* __restrict__ sums) {
  __shared__ float red[256], red2[256];
  int b = blockIdx.z, co = blockIdx.y;
  int n = blockIdx.x * 256 + threadIdx.x;
  float acc = 0.0f;
#pragma unroll 8
  for (int ci = 0; ci < ICc; ++ci)
    acc = fmaf(ww[co * ICc + ci], fxy[((size_t)b * ICc + ci) * Np + n], acc);
  wxy[((size_t)b * Cc + co) * Np + n] = acc;
  red[threadIdx.x] = acc;
  red2[threadIdx.x] = acc * acc;
  __syncthreads();
  for (int s = 128; s > 0; s >>= 1) {
    if (threadIdx.x < s) {
      red[threadIdx.x] += red[threadIdx.x + s];
      red2[threadIdx.x] += red2[threadIdx.x + s];
    }
    __syncthreads();
  }
  if (threadIdx.x == 0) {
    atomicAdd(&sums[co], red[0]);
    atomicAdd(&sums[64 + co], red2[0]);
  }
}

// --------------- kernel 5: BatchNorm (batch stats) + residual add -----------
__global__ __launch_bounds__(256)
void k_finalize(const float* __restrict__ wxy, const float* __restrict__ sums,
                const float* __restrict__ gamma, const float* __restrict__ beta,
                const float* __restrict__ x, float* __restrict__ out) {
  int idx = blockIdx.x * 256 + threadIdx.x;
  if (idx >= Bn * Cc * Np) return;
  int co = (idx / Np) % Cc;
  const float inv = 1.0f / (float)(Bn * Np);
  float mean = sums[co] * inv;
  float var = sums[64 + co] * inv - mean * mean;
  float v = (wxy[idx] - mean) * rsqrtf(var + BN_EPSf);
  out[idx] = v * gamma[co] + beta[co] + x[idx];
}

// ---------------------------------------------------------------------------
extern "C" void kernel_launch(void* const* d_in, const int* in_sizes, int n_in,
                              void* d_out, int out_size, void* d_ws, size_t ws_size,
                              hipStream_t stream) {
  const float* x       = (const float*)d_in[0];
  const float* y       = (const float*)d_in[1];
  const float* w_theta = (const float*)d_in[2];
  const float* w_phi   = (const float*)d_in[3];
  const float* w_g     = (const float*)d_in[4];
  const float* w_w     = (const float*)d_in[5];
  const float* gamma   = (const float*)d_in[6];
  const float* beta    = (const float*)d_in[7];
  float* out = (float*)d_out;

  float* ws    = (float*)d_ws;
  float* theta = ws;                                  // 4*32*9216
  float* phi   = theta + (size_t)Bn * ICc * Np;       // 4*32*2304
  float* g     = phi + (size_t)Bn * ICc * Mp;         // 4*32*2304
  float* fxy   = g + (size_t)Bn * ICc * Mp;           // 4*32*9216
  float* wxy   = fxy + (size_t)Bn * ICc * Np;         // 4*64*9216
  float* sums  = wxy + (size_t)Bn * Cc * Np;          // 128

  k_zero_sums<<<1, 128, 0, stream>>>(sums);
  k_proj_theta<<<(Bn * Np + 255) / 256, 256, 0, stream>>>(x, w_theta, theta);
  k_proj_pool<<<(Bn * Mp + 255) / 256, 256, 0, stream>>>(y, w_phi, w_g, phi, g);
  k_attention<<<dim3(Np / 16, Bn), 32, 0, stream>>>(theta, phi, g, fxy);
  k_wxy_stats<<<dim3(Np / 256, Cc, Bn), 256, 0, stream>>>(fxy, w_w, wxy, sums);
  k_finalize<<<(Bn * Cc * Np + 255) / 256, 256, 0, stream>>>(wxy, sums, gamma, beta, x, out);
}